// TwoAttentionGATOriginal_39805756900160
// MI455X (gfx1250) — compile-verified
//
#include <hip/hip_runtime.h>
#include <math.h>

typedef __attribute__((ext_vector_type(2))) float v2f;
typedef __attribute__((ext_vector_type(8))) float v8f;

#define N_NODES 20000
#define N_EDGES 160000
#define HEADS   8
#define DIM     256     // H * DV
#define DVAL    32
#define EPS_F   1e-12f
#define SLOPE_F 0.2f

// ---- monotone float <-> u32 mapping for atomic max on signed floats ----
__device__ __forceinline__ unsigned fmapu(float f) {
  unsigned u = __float_as_uint(f);
  return (u & 0x80000000u) ? ~u : (u | 0x80000000u);
}
__device__ __forceinline__ float funmap(unsigned u) {
  return (u & 0x80000000u) ? __uint_as_float(u ^ 0x80000000u)
                           : __uint_as_float(~u);
}

// ---------------------------------------------------------------------------
// Zero fill (accumulators must be re-zeroed every call: atomics accumulate)
// ---------------------------------------------------------------------------
__global__ void zero_kernel(float* __restrict__ p, size_t n) {
  size_t i = (size_t)blockIdx.x * blockDim.x + threadIdx.x;
  size_t stride = (size_t)gridDim.x * blockDim.x;
  for (; i < n; i += stride) p[i] = 0.0f;
}

// ---------------------------------------------------------------------------
// fp32 GEMM via V_WMMA_F32_16X16X4_F32.  C[M,Nc] = A[M,K] * B[K,Nc] (+bias)
// One wave per 16x16 C tile; block = 128 threads = 4 waves = 4 N-tiles.
// M % 16 == 0, Nc % 64 == 0, K % 4 == 0 (exact here, EXEC stays all-ones).
// A 16x4 layout: lanes 0-15 hold K={0,1}, lanes 16-31 hold K={2,3} (2 VGPRs).
// C/D layout: VGPR r holds M = r + 8*(lane>=16), N = lane&15.
// ---------------------------------------------------------------------------
__global__ __launch_bounds__(128)
void wmma_gemm_f32(const float* __restrict__ A, const float* __restrict__ B,
                   float* __restrict__ C, const float* __restrict__ bias,
                   int M, int K, int Nc) {
  const int lane  = threadIdx.x & 31;
  const int wave  = threadIdx.x >> 5;
  const int tileM = blockIdx.y;
  const int tileN = blockIdx.x * 4 + wave;
  const int m  = tileM * 16 + (lane & 15);
  const int n  = tileN * 16 + (lane & 15);
  const int kh = (lane >> 4) * 2;          // 0 for lanes 0-15, 2 for 16-31

  v8f acc = {};
  for (int k0 = 0; k0 < K; k0 += 4) {
    v2f a = *(const v2f*)(A + (size_t)m * K + k0 + kh);   // A[m][k0+kh .. +1]
    v2f b;
    b.x = B[(size_t)(k0 + kh)     * Nc + n];
    b.y = B[(size_t)(k0 + kh + 1) * Nc + n];
    acc = __builtin_amdgcn_wmma_f32_16x16x4_f32(
        false, a, false, b, (short)0, acc, false, false);
  }
  const int rowBase = tileM * 16 + 8 * (lane >> 4);
  const float bv = bias ? bias[n] : 0.0f;
#pragma unroll
  for (int r = 0; r < 8; ++r)
    C[(size_t)(rowBase + r) * Nc + n] = acc[r] + bv;
}

// ---------------------------------------------------------------------------
// Relation score tables: collapse (rel_inp @ W_relation) . w_rel into
//   A_tab[50][8] (deprel part) and B_tab[4][8] (deparc part).
// ---------------------------------------------------------------------------
__global__ __launch_bounds__(256)
void rel_tables_kernel(const float* __restrict__ W_relation,   // [128,128]
                       const float* __restrict__ w_rel,        // [8,16]
                       const float* __restrict__ deprel_emb,   // [50,64]
                       const float* __restrict__ deparc_emb,   // [4,64]
                       float* __restrict__ A_tab,              // [50,8]
                       float* __restrict__ B_tab) {            // [4,8]
  __shared__ float wr[128 * 8];  // wr[r][h] = sum_k W_relation[r][h*16+k]*w_rel[h][k]
  const int tid = threadIdx.x;
  for (int i = tid; i < 128 * 8; i += 256) {
    int r = i >> 3, h = i & 7;
    float s = 0.0f;
#pragma unroll
    for (int k = 0; k < 16; ++k)
      s += W_relation[r * 128 + h * 16 + k] * w_rel[h * 16 + k];
    wr[i] = s;
  }
  __syncthreads();
  for (int i = tid; i < 50 * 8; i += 256) {
    int d = i >> 3, h = i & 7;
    float s = 0.0f;
    for (int r = 0; r < 64; ++r) s += deprel_emb[d * 64 + r] * wr[r * 8 + h];
    A_tab[i] = s;
  }
  for (int i = tid; i < 4 * 8; i += 256) {
    int d = i >> 3, h = i & 7;
    float s = 0.0f;
    for (int r = 0; r < 64; ++r) s += deparc_emb[d * 64 + r] * wr[(64 + r) * 8 + h];
    B_tab[i] = s;
  }
}

// ---------------------------------------------------------------------------
// Per-node attention score tables: sv_src[n,h]=value[n,h,:].w_src[h], etc.
// ---------------------------------------------------------------------------
__global__ __launch_bounds__(256)
void node_scores_kernel(const float* __restrict__ value,
                        const float* __restrict__ w_src,
                        const float* __restrict__ w_tgt,
                        float* __restrict__ sv_src,
                        float* __restrict__ sv_tgt) {
  int i = blockIdx.x * 256 + threadIdx.x;     // n*8 + h
  if (i >= N_NODES * HEADS) return;
  int n = i >> 3, h = i & 7;
  const float* v = value + (size_t)n * DIM + h * DVAL;
  float ss = 0.0f, st = 0.0f;
#pragma unroll
  for (int d = 0; d < DVAL; ++d) {
    float x = v[d];
    ss += x * w_src[h * DVAL + d];
    st += x * w_tgt[h * DVAL + d];
  }
  sv_src[i] = ss;
  sv_tgt[i] = st;
}

// ---------------------------------------------------------------------------
// Per-edge scores + LeakyReLU, global max (block-reduced), degree count.
// ---------------------------------------------------------------------------
__global__ __launch_bounds__(256)
void edge_scores_kernel(const int* __restrict__ src, const int* __restrict__ tgt,
                        const int* __restrict__ dr, const int* __restrict__ da,
                        const float* __restrict__ sv_src,
                        const float* __restrict__ sv_tgt,
                        const float* __restrict__ A_tab,
                        const float* __restrict__ B_tab,
                        float* __restrict__ scores_v, float* __restrict__ scores_r,
                        unsigned* __restrict__ maxes,  // [0]=max_v [1]=max_r
                        float* __restrict__ deg) {
  __shared__ unsigned smv[256], smr[256];
  const int tid = threadIdx.x;
  const int i = blockIdx.x * 256 + tid;       // e*8 + h
  unsigned mv = 0u, mr = 0u;                  // mapped -inf
  if (i < N_EDGES * HEADS) {
    int e = i >> 3, h = i & 7;
    int s = src[e], t = tgt[e];
    float st = sv_tgt[t * 8 + h];
    float sv = sv_src[s * 8 + h] + st;
    sv = (sv >= 0.0f) ? sv : SLOPE_F * sv;
    float sr = A_tab[dr[e] * 8 + h] + B_tab[da[e] * 8 + h] + st;
    sr = (sr >= 0.0f) ? sr : SLOPE_F * sr;
    scores_v[i] = sv;
    scores_r[i] = sr;
    mv = fmapu(sv);
    mr = fmapu(sr);
    if (h == 0) atomicAdd(&deg[t], 1.0f);
  }
  smv[tid] = mv; smr[tid] = mr;
  __syncthreads();
  for (int s = 128; s > 0; s >>= 1) {
    if (tid < s) {
      smv[tid] = max(smv[tid], smv[tid + s]);
      smr[tid] = max(smr[tid], smr[tid + s]);
    }
    __syncthreads();
  }
  if (tid == 0) {
    atomicMax(&maxes[0], smv[0]);
    atomicMax(&maxes[1], smr[0]);
  }
}

// ---------------------------------------------------------------------------
// exp(score - global_max) in-place + scatter-add softmax denominators.
// ---------------------------------------------------------------------------
__global__ __launch_bounds__(256)
void edge_exp_kernel(float* __restrict__ scores_v, float* __restrict__ scores_r,
                     const int* __restrict__ tgt,
                     const unsigned* __restrict__ maxes,
                     float* __restrict__ denom_v, float* __restrict__ denom_r) {
  int i = blockIdx.x * 256 + threadIdx.x;
  if (i >= N_EDGES * HEADS) return;
  int e = i >> 3, h = i & 7;
  int t = tgt[e];
  float Mv = funmap(maxes[0]);
  float Mr = funmap(maxes[1]);
  float ev = expf(scores_v[i] - Mv);
  float er = expf(scores_r[i] - Mr);
  scores_v[i] = ev;
  scores_r[i] = er;
  atomicAdd(&denom_v[t * 8 + h], ev);
  atomicAdd(&denom_r[t * 8 + h], er);
}

// ---------------------------------------------------------------------------
// Per-edge gather/scatter: one block per edge, 256 threads cover (h,d).
// out_v[tgt] += value[src]*attn_v ; out_r likewise ; nb_agg[tgt] += nb[src].
// ---------------------------------------------------------------------------
__global__ __launch_bounds__(256)
void edge_scatter_kernel(const int* __restrict__ src, const int* __restrict__ tgt,
                         const float* __restrict__ value, const float* __restrict__ nb,
                         const float* __restrict__ exp_v, const float* __restrict__ exp_r,
                         const float* __restrict__ denom_v, const float* __restrict__ denom_r,
                         float* __restrict__ out_v, float* __restrict__ out_r,
                         float* __restrict__ nb_agg) {
  const int e = blockIdx.x;
  const int tid = threadIdx.x;          // h = tid>>5, d = tid&31
  const int s = src[e], t = tgt[e];
  const int h = tid >> 5;
  const float av = exp_v[e * 8 + h] / (denom_v[t * 8 + h] + EPS_F);
  const float ar = exp_r[e * 8 + h] / (denom_r[t * 8 + h] + EPS_F);
  const float x = value[(size_t)s * DIM + tid];
  atomicAdd(&out_v[(size_t)t * DIM + tid], x * av);
  atomicAdd(&out_r[(size_t)t * DIM + tid], x * ar);
  atomicAdd(&nb_agg[(size_t)t * DIM + tid], nb[(size_t)s * DIM + tid]);
}

// ---------------------------------------------------------------------------
// Gate: sigmoid([inp, nb_agg/deg] @ gate_w + gate_b) per (node, head).
// ---------------------------------------------------------------------------
__global__ __launch_bounds__(256)
void gate_kernel(const float* __restrict__ inp, const float* __restrict__ nb_agg,
                 const float* __restrict__ deg,
                 const float* __restrict__ gate_w, const float* __restrict__ gate_b,
                 float* __restrict__ gate) {
  int i = blockIdx.x * 256 + threadIdx.x;   // n*8 + h
  if (i >= N_NODES * HEADS) return;
  int n = i >> 3, h = i & 7;
  float inv = 1.0f / fmaxf(deg[n], 1.0f);
  float s = gate_b[h];
  const float* xi = inp + (size_t)n * DIM;
  const float* xa = nb_agg + (size_t)n * DIM;
  for (int k = 0; k < DIM; ++k) s += xi[k] * gate_w[k * 8 + h];
  for (int k = 0; k < DIM; ++k) s += (xa[k] * inv) * gate_w[(DIM + k) * 8 + h];
  gate[i] = 1.0f / (1.0f + expf(-s));
}

// ---------------------------------------------------------------------------
// Head projection via WMMA with implicit [out_v | out_r] concat, fused with
// gate multiply and final bias.  grid (M/16, H), block 64 = 2 waves (o-tiles).
// ---------------------------------------------------------------------------
__global__ __launch_bounds__(64)
void head_proj_wmma(const float* __restrict__ out_v, const float* __restrict__ out_r,
                    const float* __restrict__ fw,       // [8,64,32]
                    const float* __restrict__ fb,       // [8,32]
                    const float* __restrict__ gate,     // [N,8]
                    const float* __restrict__ final_bias,
                    float* __restrict__ out) {
  const int lane  = threadIdx.x & 31;
  const int wave  = threadIdx.x >> 5;
  const int h     = blockIdx.y;
  const int tileM = blockIdx.x;
  const int m  = tileM * 16 + (lane & 15);
  const int o  = wave * 16 + (lane & 15);
  const int kh = (lane >> 4) * 2;

  v8f acc = {};
#pragma unroll
  for (int k0 = 0; k0 < 64; k0 += 4) {
    int k = k0 + kh;                     // even; k and k+1 stay on same side
    const float* Asrc = (k < 32) ? out_v : out_r;
    int kk = k & 31;
    v2f a = *(const v2f*)(Asrc + (size_t)m * DIM + h * DVAL + kk);
    v2f b;
    b.x = fw[h * 2048 + k * 32 + o];
    b.y = fw[h * 2048 + (k + 1) * 32 + o];
    acc = __builtin_amdgcn_wmma_f32_16x16x4_f32(
        false, a, false, b, (short)0, acc, false, false);
  }
  const int rowBase = tileM * 16 + 8 * (lane >> 4);
  const float bo = fb[h * 32 + o];
  const float fbias = final_bias[h * 32 + o];
#pragma unroll
  for (int r = 0; r < 8; ++r) {
    int mm = rowBase + r;
    float v = (acc[r] + bo) * gate[mm * 8 + h] + fbias;
    out[(size_t)mm * DIM + h * DVAL + o] = v;
  }
}

// ---------------------------------------------------------------------------
extern "C" void kernel_launch(void* const* d_in, const int* in_sizes, int n_in,
                              void* d_out, int out_size, void* d_ws, size_t ws_size,
                              hipStream_t stream) {
  (void)in_sizes; (void)n_in; (void)out_size; (void)ws_size;

  const float* inp         = (const float*)d_in[0];
  // d_in[1] dist_edge: unused by the reference
  const float* W_value     = (const float*)d_in[2];
  const float* W_relation  = (const float*)d_in[3];
  const float* deprel_emb  = (const float*)d_in[4];
  const float* deparc_emb  = (const float*)d_in[5];
  const float* w_src       = (const float*)d_in[6];
  const float* w_tgt       = (const float*)d_in[7];
  const float* w_rel       = (const float*)d_in[8];
  const float* fw          = (const float*)d_in[9];
  const float* fb          = (const float*)d_in[10];
  const float* neighbor_w  = (const float*)d_in[11];
  const float* neighbor_b  = (const float*)d_in[12];
  const float* gate_w      = (const float*)d_in[13];
  const float* gate_b      = (const float*)d_in[14];
  const float* final_bias  = (const float*)d_in[15];
  const int*   edge_index  = (const int*)d_in[16];
  const int*   deprel_edge = (const int*)d_in[17];
  const int*   deparc_edge = (const int*)d_in[18];
  const int*   src = edge_index;
  const int*   tgt = edge_index + N_EDGES;
  float* out = (float*)d_out;

  // ---- workspace carve-out (all sub-offsets stay 8-byte aligned) ----
  float* ws = (float*)d_ws;
  size_t off = 0;
  auto carve = [&](size_t n) { float* p = ws + off; off += n; return p; };
  float* value    = carve((size_t)N_NODES * DIM);    // inp @ W_value
  float* nb       = carve((size_t)N_NODES * DIM);    // inp @ neighbor_w + b
  float* scores_v = carve((size_t)N_EDGES * HEADS);  // scores -> exp in place
  float* scores_r = carve((size_t)N_EDGES * HEADS);
  float* sv_src   = carve((size_t)N_NODES * HEADS);
  float* sv_tgt   = carve((size_t)N_NODES * HEADS);
  float* A_tab    = carve(512);
  float* B_tab    = carve(64);
  float* gate     = carve((size_t)N_NODES * HEADS);
  float* zstart   = ws + off;                        // ---- zeroed region ----
  float* out_v    = carve((size_t)N_NODES * DIM);
  float* out_r    = carve((size_t)N_NODES * DIM);
  float* nb_agg   = carve((size_t)N_NODES * DIM);
  float* denom_v  = carve((size_t)N_NODES * HEADS);
  float* denom_r  = carve((size_t)N_NODES * HEADS);
  float* deg      = carve((size_t)N_NODES);
  unsigned* maxes = (unsigned*)carve(2);             // mapped -inf == 0
  size_t zcount   = (size_t)((ws + off) - zstart);

  // 1) zero accumulators (fresh every call; atomics accumulate)
  zero_kernel<<<4096, 256, 0, stream>>>(zstart, zcount);

  // 2) dense projections via fp32 WMMA: value and neighbor features
  wmma_gemm_f32<<<dim3(DIM / 64, N_NODES / 16), 128, 0, stream>>>(
      inp, W_value, value, nullptr, N_NODES, DIM, DIM);
  wmma_gemm_f32<<<dim3(DIM / 64, N_NODES / 16), 128, 0, stream>>>(
      inp, neighbor_w, nb, neighbor_b, N_NODES, DIM, DIM);

  // 3) tiny relation tables + per-node score tables
  rel_tables_kernel<<<1, 256, 0, stream>>>(W_relation, w_rel, deprel_emb,
                                           deparc_emb, A_tab, B_tab);
  node_scores_kernel<<<(N_NODES * HEADS) / 256, 256, 0, stream>>>(
      value, w_src, w_tgt, sv_src, sv_tgt);

  // 4) edge scores + global max + degree
  edge_scores_kernel<<<(N_EDGES * HEADS) / 256, 256, 0, stream>>>(
      src, tgt, deprel_edge, deparc_edge, sv_src, sv_tgt, A_tab, B_tab,
      scores_v, scores_r, maxes, deg);

  // 5) exp + softmax denominators
  edge_exp_kernel<<<(N_EDGES * HEADS) / 256, 256, 0, stream>>>(
      scores_v, scores_r, tgt, maxes, denom_v, denom_r);

  // 6) weighted gather/scatter (the bandwidth-bound core)
  edge_scatter_kernel<<<N_EDGES, 256, 0, stream>>>(
      src, tgt, value, nb, scores_v, scores_r, denom_v, denom_r,
      out_v, out_r, nb_agg);

  // 7) gate
  gate_kernel<<<(N_NODES * HEADS) / 256, 256, 0, stream>>>(
      inp, nb_agg, deg, gate_w, gate_b, gate);

  // 8) head projection (WMMA) fused with gate and biases -> d_out
  head_proj_wmma<<<dim3(N_NODES / 16, HEADS), 64, 0, stream>>>(
      out_v, out_r, fw, fb, gate, final_bias, out);
}